// RNN_72524817760554
// MI455X (gfx1250) — compile-verified
//
#include <hip/hip_runtime.h>
#include <hip/hip_bf16.h>

// Problem constants (from reference)
#define N_B    64
#define T1_LEN 1025
#define T_STEPS 1024
#define VOCAB  10000
#define D_E    512
#define H_DIM  512
#define FOURH  2048
#define G_WG   32          // workgroups in the scan kernel (one per 16 hidden cols)

#define USE_TDM 1          // stage Wh into LDS with the Tensor Data Mover

typedef __attribute__((ext_vector_type(2))) float v2f;
typedef __attribute__((ext_vector_type(8))) float v8f;
typedef unsigned int u32x4 __attribute__((ext_vector_type(4)));
typedef unsigned int u32x8 __attribute__((ext_vector_type(8)));

// D = A(16x4 f32) * B(4x16 f32) + C(16x16 f32), full f32 precision (reference is f32;
// compute is nowhere near the matrix ceiling here, so no reason to drop precision).
__device__ __forceinline__ v8f wmma_f32_4(v2f a, v2f b, v8f c) {
  return __builtin_amdgcn_wmma_f32_16x16x4_f32(
      /*neg_a=*/false, a, /*neg_b=*/false, b,
      /*c_mod=*/(short)0, c, /*reuse_a=*/false, /*reuse_b=*/false);
}

__device__ __forceinline__ float sigmoidf_fast(float x) {
  return 1.0f / (1.0f + __expf(-x));
}

// ---------------------------------------------------------------------------
// Kernel 0: zero the cell state and the global barrier counter (ws is poisoned).
// ---------------------------------------------------------------------------
__global__ void init_state_kernel(float* __restrict__ cbuf,
                                  unsigned int* __restrict__ ctr) {
  int idx = blockIdx.x * blockDim.x + threadIdx.x;
  if (idx < N_B * H_DIM) cbuf[idx] = 0.0f;
  if (idx == 0) *ctr = 0u;
}

// ---------------------------------------------------------------------------
// Kernel 1: Ex = W_embed @ Wx + b   (10000 x 2048, f32, lives in L2 ~80MB)
// Grid: x = 2048/64 col tiles, y = ceil(10000/64) row tiles. 256 thr = 8 waves.
// Wave w: tile_m = w&3 (16 rows), col tiles 2*(w>>2), 2*(w>>2)+1.
// ---------------------------------------------------------------------------
__global__ void __launch_bounds__(256)
embed_proj_kernel(const float* __restrict__ Wemb,
                  const float* __restrict__ Wx,
                  const float* __restrict__ bias,
                  float* __restrict__ Ex) {
  const int ct   = blockIdx.x;
  const int rt   = blockIdx.y;
  const int wave = threadIdx.x >> 5;
  const int lane = threadIdx.x & 31;
  const int half = lane >> 4;
  const int l15  = lane & 15;
  const int tile_m = wave & 3;
  const int tn0    = (wave >> 2) * 2;

  const int arow   = rt * 64 + tile_m * 16 + l15;
  const int arow_c = (arow < VOCAB) ? arow : (VOCAB - 1);   // clamp, discard on store
  const int col0   = ct * 64 + tn0 * 16 + l15;
  const int col1   = col0 + 16;

  v8f acc0 = {}; v8f acc1 = {};
#pragma unroll 4
  for (int k = 0; k < D_E; k += 4) {
    const int k0 = k + 2 * half;                 // A: VGPR0 holds K={0,1}|{2,3} per half
    v2f a = *(const v2f*)(Wemb + (size_t)arow_c * D_E + k0);
    v2f b0, b1;
    b0.x = Wx[(size_t)k0 * FOURH + col0];
    b0.y = Wx[(size_t)(k0 + 1) * FOURH + col0];
    b1.x = Wx[(size_t)k0 * FOURH + col1];
    b1.y = Wx[(size_t)(k0 + 1) * FOURH + col1];
    acc0 = wmma_f32_4(a, b0, acc0);
    acc1 = wmma_f32_4(a, b1, acc1);
  }
  const float bb0 = bias[col0];
  const float bb1 = bias[col1];
#pragma unroll
  for (int r = 0; r < 8; ++r) {
    const int m  = rt * 64 + tile_m * 16 + r + 8 * half;   // C: M = r + 8*(lane>=16)
    if (m < VOCAB) {
      Ex[(size_t)m * FOURH + col0] = acc0[r] + bb0;
      Ex[(size_t)m * FOURH + col1] = acc1[r] + bb1;
    }
  }
}

// ---------------------------------------------------------------------------
// Kernel 2: persistent LSTM scan. 32 WGs x 256 threads. WG g owns hidden cols
// [16g, 16g+16) and their 4 gate column blocks at {0,512,1024,1536}+16g.
// Wh slice (512x64 f32) is staged into LDS ONCE via the Tensor Data Mover and
// reused for all 1024 steps. h(t) is written to d_out[:, t, :] and read back
// next step, synchronized with a monotone device-scope atomic barrier.
// ---------------------------------------------------------------------------
#define WH_STRIDE 72   // 512x72 f32 = 144KB; halves of the wave hit disjoint banks
#define A_STRIDE  65   // 64x65 f32 ~ 16.6KB staging for gate pre-activations

__global__ void __launch_bounds__(256)
lstm_scan_kernel(const int* __restrict__ captions,    // (N, T1) tokens
                 const float* __restrict__ h_init,    // (N, H)
                 const float* __restrict__ Wh,        // (H, 4H)
                 const float* __restrict__ Ex,        // (V, 4H) = Wemb@Wx + b
                 float* __restrict__ cbuf,            // (N, H) cell state
                 unsigned int* __restrict__ ctr,      // global barrier counter
                 float* __restrict__ out) {           // (N, T, H)
  extern __shared__ float lds[];
  float* ldsWh = lds;                        // [512][WH_STRIDE]
  float* ldsA  = lds + 512 * WH_STRIDE;      // [64][A_STRIDE]

  const int wg   = blockIdx.x;               // hidden block 0..31
  const int tid  = threadIdx.x;
  const int wave = tid >> 5;
  const int lane = tid & 31;
  const int half = lane >> 4;
  const int l15  = lane & 15;
  const int tile_m = wave & 3;               // 16-row batch tile
  const int q0     = (wave >> 2) * 2;        // gate pair {0,1} or {2,3}

  // ---- stage this WG's Wh slice into LDS (once) ----
#if USE_TDM
  // 4 TDM tile loads (one per gate block): each is a 16-wide x 512-row f32 tile
  // of the (512 x 2048) Wh tensor. TDM hardware padding (16 DWORDs data, then
  // +56 DWORDs skip) lays rows out with a 72-float stride; the 4 tiles start at
  // column offsets 0/16/32/48, interleaving into the [512][72] LDS image.
  if (wave == 0) {
    const unsigned long long whBase = (unsigned long long)(size_t)Wh;
    const unsigned ldsBase = (unsigned)(size_t)ldsWh;    // LDS byte offset
#pragma unroll
    for (int q = 0; q < 4; ++q) {
      const unsigned long long gaddr =
          whBase + (unsigned long long)((q * H_DIM + wg * 16) * 4);
      u32x4 g0;
      g0[0] = 1u;                                        // count=1 (valid user D#)
      g0[1] = ldsBase + (unsigned)(q * 16 * 4);          // lds_addr: tile base
      g0[2] = (unsigned)(gaddr & 0xFFFFFFFFull);         // global_addr[31:0]
      g0[3] = (unsigned)((gaddr >> 32) & 0x01FFFFFFull)  // global_addr[56:32]
              | (2u << 30);                              // type=2 ("image")
      u32x8 g1;
      g1[0] = (2u << 16)      // data_size = 4 bytes
            | (1u << 20)      // pad_enable
            | (3u << 22)      // pad_interval: 16 DWORDs of data
            | (55u << 25);    // pad_amount: 56 DWORDs -> 72-float row stride
      g1[1] = (2048u << 16);  // tensor_dim0[15:0] = 2048 (atomic_barrier_addr=0)
      g1[2] = (512u << 16);   // tensor_dim0[31:16]=0 | tensor_dim1[15:0] = 512
      g1[3] = (16u << 16);    // tensor_dim1[31:16]=0 | tile_dim0 = 16
      g1[4] = 512u;           // tile_dim1 = 512 | tile_dim2 = 0
      g1[5] = 2048u;          // tensor_dim0_stride[31:0] = 2048 elements
      g1[6] = 0u;             // stride hi | tensor_dim1_stride lo (unused, 2D)
      g1[7] = 0u;
      u32x4 g2 = {0u, 0u, 0u, 0u};   // dims 2+ unused (tile_dim3 = 0)
      u32x4 g3 = {0u, 0u, 0u, 0u};
      asm volatile("tensor_load_to_lds %0, %1, %2, %3"
                   :: "s"(g0), "s"(g1), "s"(g2), "s"(g3)
                   : "memory");
    }
    __builtin_amdgcn_s_wait_tensorcnt(0);
  }
#else
  for (int idx = tid; idx < 512 * 64; idx += 256) {
    const int k = idx >> 6;
    const int c = idx & 63;                                  // local col 0..63
    const int gc = (c >> 4) * H_DIM + wg * 16 + (c & 15);    // global gate col
    ldsWh[k * WH_STRIDE + c] = Wh[(size_t)k * FOURH + gc];
  }
#endif
  __syncthreads();

  const int mrow = tile_m * 16 + l15;        // batch row for A frag (0..63)
  const int cq0  = q0 * 16 + l15;            // local B/C col, gate q0
  const int cq1  = cq0 + 16;                 // local B/C col, gate q0+1

  for (int t = 0; t < T_STEPS; ++t) {
    // h source: h_init at t==0, else previous step's slab of the output tensor
    const float* hsrc = (t == 0) ? h_init : (out + (size_t)(t - 1) * H_DIM);
    const size_t hstr = (t == 0) ? (size_t)H_DIM : (size_t)T_STEPS * H_DIM;

    // ---- a_tile = h @ Wh[:, gate cols]  (two 16x16 accumulators per wave) ----
    v8f acc0 = {}; v8f acc1 = {};
#pragma unroll 4
    for (int k = 0; k < H_DIM; k += 4) {
      const int k0 = k + 2 * half;
      v2f a = *(const v2f*)(hsrc + (size_t)mrow * hstr + k0);
      v2f b0, b1;
      b0.x = ldsWh[k0 * WH_STRIDE + cq0];
      b0.y = ldsWh[(k0 + 1) * WH_STRIDE + cq0];
      b1.x = ldsWh[k0 * WH_STRIDE + cq1];
      b1.y = ldsWh[(k0 + 1) * WH_STRIDE + cq1];
      acc0 = wmma_f32_4(a, b0, acc0);
      acc1 = wmma_f32_4(a, b1, acc1);
    }

    // ---- add Ex[token] gather (fused embed+input-projection), park in LDS ----
#pragma unroll
    for (int r = 0; r < 8; ++r) {
      const int m   = tile_m * 16 + r + 8 * half;            // batch row of C elem
      const int tok = captions[m * T1_LEN + t];              // captions[:, :-1]
      const int g0c = q0 * H_DIM + wg * 16 + l15;            // global gate col, gate q0
      ldsA[m * A_STRIDE + cq0] = acc0[r] + Ex[(size_t)tok * FOURH + g0c];
      ldsA[m * A_STRIDE + cq1] = acc1[r] + Ex[(size_t)tok * FOURH + g0c + H_DIM];
    }
    __syncthreads();

    // ---- gates + state update: 64 rows x 16 hidden cols = 1024 elems ----
#pragma unroll
    for (int e = 0; e < 4; ++e) {
      const int idx = tid + e * 256;
      const int n = idx >> 4;
      const int j = idx & 15;
      const float ai = ldsA[n * A_STRIDE + j];
      const float af = ldsA[n * A_STRIDE + 16 + j];
      const float ao = ldsA[n * A_STRIDE + 32 + j];
      const float ag = ldsA[n * A_STRIDE + 48 + j];
      const int hc = wg * 16 + j;
      const float c_old = cbuf[n * H_DIM + hc];
      const float c_new = sigmoidf_fast(af) * c_old +
                          sigmoidf_fast(ai) * tanhf(ag);
      cbuf[n * H_DIM + hc] = c_new;
      out[(size_t)n * T_STEPS * H_DIM + (size_t)t * H_DIM + hc] =
          sigmoidf_fast(ao) * tanhf(c_new);
    }

    // ---- device-scope barrier: everyone's h(t) visible before step t+1 ----
    __threadfence();            // release h stores to device scope
    __syncthreads();            // LDS a-tile reuse guard + all lanes arrived
    if (tid == 0) {
      __hip_atomic_fetch_add(ctr, 1u, __ATOMIC_RELEASE, __HIP_MEMORY_SCOPE_AGENT);
      const unsigned target = (unsigned)(t + 1) * G_WG;     // monotone, no reset race
      while (__hip_atomic_load(ctr, __ATOMIC_ACQUIRE, __HIP_MEMORY_SCOPE_AGENT) < target) {
        __builtin_amdgcn_s_sleep(4);
      }
    }
    __syncthreads();
    __threadfence();            // acquire: invalidate stale h lines in WGP$
  }
}

// ---------------------------------------------------------------------------
extern "C" void kernel_launch(void* const* d_in, const int* in_sizes, int n_in,
                              void* d_out, int out_size, void* d_ws, size_t ws_size,
                              hipStream_t stream) {
  const int*   captions = (const int*)d_in[0];   // (N, T1)
  const float* h_init   = (const float*)d_in[1]; // (N, H)
  const float* W_embed  = (const float*)d_in[2]; // (V, D)
  const float* Wx       = (const float*)d_in[3]; // (D, 4H)
  const float* Wh       = (const float*)d_in[4]; // (H, 4H)
  const float* b        = (const float*)d_in[5]; // (4H,)
  float* out = (float*)d_out;                    // (N, T, H)

  // workspace layout: Ex table (80MB) | cell state (128KB) | barrier counter
  float* Ex = (float*)d_ws;
  const size_t exBytes = (size_t)VOCAB * FOURH * sizeof(float);
  float* cbuf = (float*)((char*)d_ws + exBytes);
  unsigned int* ctr =
      (unsigned int*)((char*)d_ws + exBytes + (size_t)N_B * H_DIM * sizeof(float));

  init_state_kernel<<<(N_B * H_DIM + 255) / 256, 256, 0, stream>>>(cbuf, ctr);

  dim3 egrid(FOURH / 64, (VOCAB + 63) / 64);
  embed_proj_kernel<<<egrid, 256, 0, stream>>>(W_embed, Wx, b, Ex);

  const size_t ldsBytes = (size_t)(512 * WH_STRIDE + 64 * A_STRIDE) * sizeof(float);
  lstm_scan_kernel<<<G_WG, 256, ldsBytes, stream>>>(captions, h_init, Wh, Ex,
                                                    cbuf, ctr, out);
}